// CriticGCN_601295422145
// MI455X (gfx1250) — compile-verified
//
#include <hip/hip_runtime.h>

// CriticGCN collapsed kernel for MI455X (gfx1250, wave32).
//
// out[n] = agg[n]*s1 + s2 where
//   s1 = sum_h W1[h]*W2[h],  s2 = sum_h b1[h]*W2[h] + b2
//   agg = GCN symmetric-normalized scalar aggregation.
//
// Edge passes are the bottleneck (~96MB @ 23.3TB/s ~= 4us floor + L2 atomics);
// all gather tables (1MB each) are L2-resident. The H=1024 dot products are
// done with V_WMMA_F32_16X16X4_F32 trace accumulation in one wave.

typedef __attribute__((ext_vector_type(2))) float v2f;
typedef __attribute__((ext_vector_type(8))) float v8f;

// ---------------- node kernels ----------------

__global__ void zero_cnt_kernel(unsigned* __restrict__ cnt, int n) {
  int i = blockIdx.x * blockDim.x + threadIdx.x;
  if (i < n) cnt[i] = 0u;
}

// deg counting over dst indices, 4 edges / thread
__global__ void count_deg_kernel(const int* __restrict__ dst,
                                 unsigned* __restrict__ cnt, int e4) {
  int i = blockIdx.x * blockDim.x + threadIdx.x;
  if (i < e4) {
    int4 d = ((const int4*)dst)[i];
    atomicAdd(&cnt[d.x], 1u);
    atomicAdd(&cnt[d.y], 1u);
    atomicAdd(&cnt[d.z], 1u);
    atomicAdd(&cnt[d.w], 1u);
  }
}

__global__ void count_deg_tail_kernel(const int* __restrict__ dst,
                                      unsigned* __restrict__ cnt, int n) {
  int i = blockIdx.x * blockDim.x + threadIdx.x;
  if (i < n) atomicAdd(&cnt[dst[i]], 1u);
}

// cnt (aliased in agg buffer) -> dinv, sd = dinv*state, agg = dinv^2*state (self loop)
__global__ void node_prep_kernel(const float* __restrict__ state,
                                 float* agg,      // aliased: holds u32 counts on entry
                                 float* __restrict__ dinv,
                                 float* __restrict__ sd, int n) {
  int i = blockIdx.x * blockDim.x + threadIdx.x;
  if (i < n) {
    unsigned c = ((const unsigned*)agg)[i];
    float di = rsqrtf((float)c + 1.0f);
    float s = state[i];
    dinv[i] = di;
    sd[i]   = di * s;
    agg[i]  = di * di * s;
  }
}

// ---------------- edge scatter ----------------

__global__ void scatter_kernel(const int* __restrict__ src,
                               const int* __restrict__ dst,
                               const float* __restrict__ dinv,
                               const float* __restrict__ sd,
                               float* __restrict__ agg, int e4) {
  int i = blockIdx.x * blockDim.x + threadIdx.x;
  if (i < e4) {
    int4 s4 = ((const int4*)src)[i];
    int4 d4 = ((const int4*)dst)[i];
    atomicAdd(&agg[d4.x], sd[s4.x] * dinv[d4.x]);
    atomicAdd(&agg[d4.y], sd[s4.y] * dinv[d4.y]);
    atomicAdd(&agg[d4.z], sd[s4.z] * dinv[d4.z]);
    atomicAdd(&agg[d4.w], sd[s4.w] * dinv[d4.w]);
  }
}

__global__ void scatter_tail_kernel(const int* __restrict__ src,
                                    const int* __restrict__ dst,
                                    const float* __restrict__ dinv,
                                    const float* __restrict__ sd,
                                    float* __restrict__ agg, int n) {
  int i = blockIdx.x * blockDim.x + threadIdx.x;
  if (i < n) {
    int s = src[i], d = dst[i];
    atomicAdd(&agg[d], sd[s] * dinv[d]);
  }
}

// ---------------- WMMA dot products ----------------
// One wave32. Per 64-element chunk: A(16x4)=W1 chunk, B(4x16)=W2 chunk laid out
// with identical per-lane slot index -> trace(D) = 64-elem dot. Accumulate C over
// H/64 chunks, then trace-extract (lanes 0..7 give D[v][v], lanes 24..31 give
// D[v+8][v+8] from VGPR v) and reduce through LDS.
__global__ void __launch_bounds__(32)
wmma_dot_kernel(const float* __restrict__ W1, const float* __restrict__ b1,
                const float* __restrict__ W2, const float* __restrict__ b2,
                float* __restrict__ s_out, int H) {
  int lane = threadIdx.x;                       // 0..31
  int idx  = 4 * (lane & 15) + 2 * (lane >> 4); // shared A/B slot index
  v8f c1 = {};  // accumulates W1 . W2
  v8f c2 = {};  // accumulates b1 . W2
  for (int base = 0; base < H; base += 64) {
    v2f a1 = *(const v2f*)(W1 + base + idx);
    v2f a2 = *(const v2f*)(b1 + base + idx);
    v2f bb = *(const v2f*)(W2 + base + idx);
    c1 = __builtin_amdgcn_wmma_f32_16x16x4_f32(false, a1, false, bb,
                                               (short)0, c1, false, false);
    c2 = __builtin_amdgcn_wmma_f32_16x16x4_f32(false, a2, false, bb,
                                               (short)0, c2, false, false);
  }
  union { v8f v; float f[8]; } u1, u2;
  u1.v = c1; u2.v = c2;
  float t1 = 0.0f, t2 = 0.0f;
  if (lane < 8)        { t1 = u1.f[lane];      t2 = u2.f[lane];      }
  else if (lane >= 24) { t1 = u1.f[lane - 24]; t2 = u2.f[lane - 24]; }
  __shared__ float r1[32], r2[32];
  r1[lane] = t1; r2[lane] = t2;
  __syncthreads();
  if (lane == 0) {
    float a = 0.0f, b = 0.0f;
    for (int k = 0; k < 32; ++k) { a += r1[k]; b += r2[k]; }
    s_out[0] = a;
    s_out[1] = b + b2[0];
  }
}

// ---------------- finalize ----------------

__global__ void finalize_kernel(const float* __restrict__ agg,
                                const float* __restrict__ s,
                                float* __restrict__ out, int n) {
  int i = blockIdx.x * blockDim.x + threadIdx.x;
  if (i < n) out[i] = fmaf(agg[i], s[0], s[1]);
}

// ---------------- launch ----------------

extern "C" void kernel_launch(void* const* d_in, const int* in_sizes, int n_in,
                              void* d_out, int out_size, void* d_ws, size_t ws_size,
                              hipStream_t stream) {
  (void)n_in; (void)out_size; (void)ws_size;
  const float* state = (const float*)d_in[0];
  const int*   eidx  = (const int*)d_in[1];
  const float* W1    = (const float*)d_in[2];
  const float* b1    = (const float*)d_in[3];
  const float* W2    = (const float*)d_in[4];
  const float* b2    = (const float*)d_in[5];
  float* out = (float*)d_out;

  const int N = in_sizes[0];
  const int E = in_sizes[1] / 2;
  const int H = in_sizes[2];

  const int* src = eidx;        // edge_index row 0
  const int* dst = eidx + E;    // edge_index row 1

  float* ws   = (float*)d_ws;
  float* agg  = ws;             // N floats (holds u32 counts first)
  float* dinv = ws + (size_t)N; // N floats
  float* sd   = ws + 2 * (size_t)N; // N floats
  float* scal = ws + 3 * (size_t)N; // s1, s2

  const int BT = 256;
  const int nb_n = (N + BT - 1) / BT;
  const int e4 = E / 4, rem = E % 4;
  const int nb_e4 = (e4 + BT - 1) / BT;

  zero_cnt_kernel<<<nb_n, BT, 0, stream>>>((unsigned*)agg, N);
  if (e4) count_deg_kernel<<<nb_e4, BT, 0, stream>>>(dst, (unsigned*)agg, e4);
  if (rem) count_deg_tail_kernel<<<1, BT, 0, stream>>>(dst + 4 * e4, (unsigned*)agg, rem);
  node_prep_kernel<<<nb_n, BT, 0, stream>>>(state, agg, dinv, sd, N);
  wmma_dot_kernel<<<1, 32, 0, stream>>>(W1, b1, W2, b2, scal, H);
  if (e4) scatter_kernel<<<nb_e4, BT, 0, stream>>>(src, dst, dinv, sd, agg, e4);
  if (rem) scatter_tail_kernel<<<1, BT, 0, stream>>>(src + 4 * e4, dst + 4 * e4,
                                                     dinv, sd, agg, rem);
  finalize_kernel<<<nb_n, BT, 0, stream>>>(agg, scal, out, N);
}